// EdgewiseEnergySumSEGNN_64080912056845
// MI455X (gfx1250) — compile-verified
//
#include <hip/hip_runtime.h>
#include <stdint.h>

// ---------------------------------------------------------------------------
// EdgewiseEnergySumSEGNN for MI455X (gfx1250)
//
// atom_eng[c] = FACTOR * sum_{edges e with center c} edge_eng[e] *
//               per_edge_scales[species[c], species[neighbor(e)]]
//
// Bandwidth-bound scatter-add: ~77 MB streamed once, 6.4M f32 atomics into a
// 400 KB L2-resident accumulator.  No matmul -> no WMMA; instead we use the
// CDNA5 async global->LDS copy (ASYNCcnt path) for the scale table, NT-hinted
// b128 streaming loads, and no-return device-scope global_atomic_add_f32.
// ---------------------------------------------------------------------------

typedef __attribute__((ext_vector_type(4))) float v4f;
typedef __attribute__((ext_vector_type(4))) int   v4i;

#define FACTOR 0.125f  // 1/sqrt(64), exact in fp32

// No-return f32 atomic add, device scope (resolved at the global L2 atomic
// units).  Inline asm guarantees the HW instruction (no CAS-loop fallback)
// and avoids the return-data path entirely (tracked by STOREcnt; S_ENDPGM
// performs an implicit wait-idle).
__device__ __forceinline__ void global_fadd_f32(float* p, float v) {
  asm volatile("global_atomic_add_f32 %0, %1, off scope:SCOPE_DEV"
               :: "v"(p), "v"(v)
               : "memory");
}

__global__ void __launch_bounds__(256)
edgewise_zero_kernel(float* __restrict__ out, int n) {
  int i = blockIdx.x * blockDim.x + threadIdx.x;
  if (i < n) out[i] = 0.0f;
}

__global__ void __launch_bounds__(256)
edgewise_scatter_kernel(const float* __restrict__ edge_eng,   // [E]
                        const float* __restrict__ scales,     // [16*16]
                        const int*   __restrict__ centers,    // [E]
                        const int*   __restrict__ neighbors,  // [E]
                        const int*   __restrict__ species,    // [N]
                        float*       __restrict__ out,        // [N]
                        int E) {
  __shared__ float tbl[256];

  // --- Stage the 16x16 scale table into LDS via CDNA5 async global->LDS ---
  // Each of the 256 threads moves one dword directly into LDS (no VGPR
  // round-trip).  Low 32 bits of a generic shared pointer are the LDS byte
  // offset the DS/async units expect.
  {
    unsigned tid     = threadIdx.x;
    unsigned lds_off = (unsigned)(uintptr_t)(&tbl[tid]);
    unsigned voff    = tid * 4u;
    asm volatile("global_load_async_to_lds_b32 %0, %1, %2"
                 :: "v"(lds_off), "v"(voff), "s"(scales)
                 : "memory");
    asm volatile("s_wait_asynccnt 0x0" ::: "memory");
    __syncthreads();
    // Fold the 1/sqrt(avg_neighbors) normalization into the table once per
    // block: saves one VALU mul per edge in the hot loop.
    tbl[tid] *= FACTOR;
    __syncthreads();
  }

  const int gsize = gridDim.x * blockDim.x;
  const int gtid  = blockIdx.x * blockDim.x + threadIdx.x;

  // Vector (4-edges-per-iteration) path only when E is quad-divisible so the
  // neighbors row (edge_index + E) stays 16B-aligned.
  const int nq = ((E & 3) == 0) ? (E >> 2) : 0;

  const v4f* __restrict__ ee4 = (const v4f*)edge_eng;
  const v4i* __restrict__ cc4 = (const v4i*)centers;
  const v4i* __restrict__ nn4 = (const v4i*)neighbors;

  for (int q = gtid; q < nq; q += gsize) {
    // Streamed-once data: non-temporal so the 77 MB stream does not thrash
    // the L2 lines holding species[] and the atomic accumulator.
    v4i c = __builtin_nontemporal_load(&cc4[q]);
    v4i n = __builtin_nontemporal_load(&nn4[q]);
    v4f e = __builtin_nontemporal_load(&ee4[q]);

    int i0 = species[c.x] * 16 + species[n.x];
    int i1 = species[c.y] * 16 + species[n.y];
    int i2 = species[c.z] * 16 + species[n.z];
    int i3 = species[c.w] * 16 + species[n.w];

    global_fadd_f32(out + c.x, e.x * tbl[i0]);
    global_fadd_f32(out + c.y, e.y * tbl[i1]);
    global_fadd_f32(out + c.z, e.z * tbl[i2]);
    global_fadd_f32(out + c.w, e.w * tbl[i3]);
  }

  // Scalar tail (also the full path if E % 4 != 0).
  for (int i = (nq << 2) + gtid; i < E; i += gsize) {
    int c = centers[i];
    int n = neighbors[i];
    float v = edge_eng[i] * tbl[species[c] * 16 + species[n]];
    global_fadd_f32(out + c, v);
  }
}

extern "C" void kernel_launch(void* const* d_in, const int* in_sizes, int n_in,
                              void* d_out, int out_size, void* d_ws, size_t ws_size,
                              hipStream_t stream) {
  (void)n_in; (void)d_ws; (void)ws_size;

  const float* edge_eng = (const float*)d_in[0];  // [E]
  const float* scales   = (const float*)d_in[1];  // [16,16]
  const int*   eidx     = (const int*)d_in[2];    // [2,E] row-major
  const int*   species  = (const int*)d_in[3];    // [N]
  float*       out      = (float*)d_out;          // [N,1] fp32

  const int E = in_sizes[0];
  const int* centers   = eidx;
  const int* neighbors = eidx + E;

  // Atomics accumulate -> output must be zeroed every call (graph replays).
  int zblocks = (out_size + 255) / 256;
  if (zblocks < 1) zblocks = 1;
  edgewise_zero_kernel<<<zblocks, 256, 0, stream>>>(out, out_size);

  int nq = E >> 2;
  int blocks = (nq + 255) / 256;
  if (blocks > 4096) blocks = 4096;
  if (blocks < 1) blocks = 1;
  edgewise_scatter_kernel<<<blocks, 256, 0, stream>>>(
      edge_eng, scales, centers, neighbors, species, out, E);
}